// ChargeModel_9543417332339
// MI455X (gfx1250) — compile-verified
//
#include <hip/hip_runtime.h>
#include <hip/hip_bf16.h>

typedef __attribute__((ext_vector_type(2))) float v2f;
typedef __attribute__((ext_vector_type(8))) float v8f;

#define TPB 256

// ---------------------------------------------------------------- init
// deg starts at 1.0 (self-loop weight); per-graph sums/cnt zeroed.
__global__ void k_init(float* __restrict__ deg, float* __restrict__ sums,
                       float* __restrict__ cnt, int N, int G) {
    int i = blockIdx.x * blockDim.x + threadIdx.x;
    if (i < N) deg[i] = 1.0f;
    if (i < G) { sums[i] = 0.0f; cnt[i] = 0.0f; }
}

// ---------------------------------------------------------------- pass 1: degree
__global__ void k_deg(const int* __restrict__ dst, const float* __restrict__ w,
                      float* __restrict__ deg, int E) {
    int e = blockIdx.x * blockDim.x + threadIdx.x;
    if (e < E) atomicAdd(deg + dst[e], w[e]);
}

// ---------------------------------------------------------------- dinv + agg1 self loop
__global__ void k_dinv(const float* __restrict__ x, float* __restrict__ deg_dinv,
                       float* __restrict__ agg1, int N) {
    int i = blockIdx.x * blockDim.x + threadIdx.x;
    if (i < N) {
        float d  = deg_dinv[i];
        float di = d > 0.0f ? rsqrtf(d) : 0.0f;
        deg_dinv[i] = di;                 // reuse buffer as dinv
        agg1[i]     = x[i] * di * di;     // self-loop contribution
    }
}

// ---------------------------------------------------------------- edge scatter (used for both layers)
__global__ void k_edge_scatter(const int* __restrict__ src, const int* __restrict__ dst,
                               const float* __restrict__ w, const float* __restrict__ feat,
                               const float* __restrict__ dinv, float* __restrict__ acc, int E) {
    int e = blockIdx.x * blockDim.x + threadIdx.x;
    if (e < E) {
        int s = src[e], d = dst[e];
        atomicAdd(acc + d, feat[s] * dinv[s] * w[e] * dinv[d]);
    }
}

// ---------------------------------------------------------------- dense MLP stage via WMMA f32
// z[n] = sum_k W2[k] * relu(W1[k]*agg1[n] + b1[k]);  f3[n] = z[n]*dinv[n]^2 (layer-2 self loop)
// One wave handles 16 nodes: D(16x16) = sum_{j=0..15} A_j(16x4) x B_j(4x16),
// A[m,k]=relu(W1*agg+b1), B[k,0]=W2[k] (other columns zeroed by mask-multiply,
// NOT a ternary -> avoids divergent predicated loads / EXEC churn).
__global__ void k_mlp_wmma(const float* __restrict__ agg1, const float* __restrict__ dinv,
                           const float* __restrict__ W1, const float* __restrict__ b1,
                           const float* __restrict__ W2,
                           float* __restrict__ z, float* __restrict__ f3, int N) {
    int wave = (blockIdx.x * blockDim.x + threadIdx.x) >> 5;
    int lane = threadIdx.x & 31;
    int base = wave << 4;               // 16 nodes per wave
    if (base >= N) return;              // wave-uniform exit; EXEC stays all-ones below

    int m     = lane & 15;              // A: row M; B: column N
    int node  = base + m;
    int nodec = node < N ? node : N - 1;   // clamp keeps EXEC full on tail tiles
    float a   = agg1[nodec];
    int khalf = (lane >> 4) << 1;       // lanes 0-15 hold K={0,1}, lanes 16-31 K={2,3}
    float cm  = (m == 0) ? 1.0f : 0.0f; // column-0 mask, computed once

    v8f acc = {};
    #pragma unroll
    for (int j = 0; j < 16; ++j) {
        int k0 = 4 * j + khalf;
        int k1 = k0 + 1;
        float w1a = W1[k0], w1b = W1[k1];   // unconditional loads (b64-pairable)
        float bb0 = b1[k0], bb1 = b1[k1];
        float w2a = W2[k0], w2b = W2[k1];
        v2f av, bv;
        av.x = fmaxf(w1a * a + bb0, 0.0f);
        av.y = fmaxf(w1b * a + bb1, 0.0f);
        bv.x = w2a * cm;                    // mask-multiply, no branches
        bv.y = w2b * cm;
        acc = __builtin_amdgcn_wmma_f32_16x16x4_f32(
            /*neg_a=*/false, av, /*neg_b=*/false, bv,
            /*c_mod=*/(short)0, acc, /*reuse_a=*/false, /*reuse_b=*/false);
    }

    // D column 0: lane 0 holds M=0..7 in acc[0..7]; lane 16 holds M=8..15.
    int rowbase = (lane == 0) ? base : ((lane == 16) ? base + 8 : -1);
    if (rowbase >= 0) {
        #pragma unroll
        for (int r = 0; r < 8; ++r) {
            int n2 = rowbase + r;
            if (n2 < N) {
                float zz = acc[r];
                z[n2] = zz;
                float di = dinv[n2];
                f3[n2] = zz * di * di;
            }
        }
    }
}

// ---------------------------------------------------------------- per-graph pooling
__global__ void k_pool(const float* __restrict__ f3, const float* __restrict__ b2,
                       const int* __restrict__ batch, float* __restrict__ sums,
                       float* __restrict__ cnt, int N) {
    int i = blockIdx.x * blockDim.x + threadIdx.x;
    if (i < N) {
        float v = f3[i] + b2[0];
        int g = batch[i];
        atomicAdd(sums + g, v);
        atomicAdd(cnt + g, 1.0f);
    }
}

__global__ void k_finalize(const float* __restrict__ sums, const float* __restrict__ cnt,
                           float* __restrict__ out, int G) {
    int g = blockIdx.x * blockDim.x + threadIdx.x;
    if (g < G) out[g] = sums[g] / fmaxf(cnt[g], 1.0f);
}

extern "C" void kernel_launch(void* const* d_in, const int* in_sizes, int n_in,
                              void* d_out, int out_size, void* d_ws, size_t ws_size,
                              hipStream_t stream) {
    const float* x     = (const float*)d_in[0];
    const int*   ei    = (const int*)  d_in[1];   // (2,E): row0=src, row1=dst
    const float* ew    = (const float*)d_in[2];
    const int*   batch = (const int*)  d_in[3];
    // d_in[4] = n_graphs scalar (G == out_size, unused)
    const float* W1    = (const float*)d_in[5];
    const float* b1    = (const float*)d_in[6];
    const float* W2    = (const float*)d_in[7];
    const float* b2    = (const float*)d_in[8];

    const int N = in_sizes[0];
    const int E = in_sizes[2];
    const int G = out_size;

    const int* src = ei;
    const int* dst = ei + E;

    float* ws   = (float*)d_ws;
    float* deg  = ws;            // later holds dinv
    float* agg1 = ws + (size_t)N;
    float* z    = ws + (size_t)2 * N;
    float* f3   = ws + (size_t)3 * N;   // layer-2 output accumulator
    float* sums = ws + (size_t)4 * N;
    float* cnt  = sums + G;

    float* out = (float*)d_out;

    int nbNG = (max(N, G) + TPB - 1) / TPB;
    int nbN  = (N + TPB - 1) / TPB;
    int nbE  = (E + TPB - 1) / TPB;
    int nbG  = (G + TPB - 1) / TPB;
    int waves = (N + 15) / 16;
    int nbW  = (waves * 32 + TPB - 1) / TPB;

    k_init<<<nbNG, TPB, 0, stream>>>(deg, sums, cnt, N, G);
    k_deg<<<nbE, TPB, 0, stream>>>(dst, ew, deg, E);
    k_dinv<<<nbN, TPB, 0, stream>>>(x, deg, agg1, N);
    k_edge_scatter<<<nbE, TPB, 0, stream>>>(src, dst, ew, x, deg, agg1, E);
    k_mlp_wmma<<<nbW, TPB, 0, stream>>>(agg1, deg, W1, b1, W2, z, f3, N);
    k_edge_scatter<<<nbE, TPB, 0, stream>>>(src, dst, ew, z, deg, f3, E);
    k_pool<<<nbN, TPB, 0, stream>>>(f3, b2, batch, sums, cnt, N);
    k_finalize<<<nbG, TPB, 0, stream>>>(sums, cnt, out, G);
}